// KANLayer_11218454577795
// MI455X (gfx1250) — compile-verified
//
#include <hip/hip_runtime.h>

typedef __attribute__((ext_vector_type(2))) float v2f;
typedef __attribute__((ext_vector_type(4))) float v4f;
typedef __attribute__((ext_vector_type(8))) float v8f;

#define IN_F            1024
#define ROWS_PER_WAVE   16
#define WAVES_PER_BLOCK 8
#define CHUNK           64
#define NCHUNK          (IN_F / CHUNK)               // 16
#define LDS_STRIDE      (CHUNK + 4)                  // 68 floats: conflict-free banks, rows 16B-aligned (272B)
#define WBUF            (ROWS_PER_WAVE * LDS_STRIDE) // floats per LDS buffer per wave

// out[b, :] = broadcast( sum_k x[b, k] ), computed as x_tile(16x4) @ ones(4x16)
// accumulated with V_WMMA_F32_16X16X4_F32 -> every column of C is the row sum.
// Each wave owns 16 rows and a private double-buffered LDS slice (same-wave DS
// ops are in-order, so no barriers). Fully unrolled so the staging registers
// stay in VGPRs (no scratch spill) and chunk i+1's global loads overlap the
// WMMAs of chunk i. Global traffic is pure streaming -> non-temporal hints.
__global__ __launch_bounds__(256) void kan_rowsum_wmma(const float* __restrict__ x,
                                                       float* __restrict__ out) {
    __shared__ float sm[WAVES_PER_BLOCK * 2 * WBUF];

    const int lane = threadIdx.x & 31;
    const int wid  = threadIdx.x >> 5;
    const size_t r0 = ((size_t)blockIdx.x * WAVES_PER_BLOCK + wid) * ROWS_PER_WAVE;
    float* w0 = &sm[wid * 2 * WBUF];
    float* w1 = w0 + WBUF;

    // Staging map: pass `it` covers row {2*it + lane/16}, cols (lane&15)*4.
    const int scol  = (lane & 15) << 2;
    const int srow0 = lane >> 4;
    // A-operand feed (32-bit A 16x4): lane&15 = M row,
    // lanes 0-15 supply K = {k, k+1}, lanes 16-31 supply K = {k+2, k+3}.
    const int rowA  = lane & 15;
    const int kbase = (lane >> 4) << 1;

    const float* gbase = x + r0 * (size_t)IN_F + scol;

    v2f bones;
    bones[0] = 1.0f;
    bones[1] = 1.0f;
    v8f c = {};
    v4f d[8];

    // Prologue: chunk 0 -> buffer 0 (8 NT b128 loads in flight, then drain to LDS).
    #pragma unroll
    for (int it = 0; it < 8; ++it)
        d[it] = __builtin_nontemporal_load(
                    (const v4f*)(gbase + (size_t)(it * 2 + srow0) * IN_F));
    #pragma unroll
    for (int it = 0; it < 8; ++it)
        *(v4f*)&w0[(it * 2 + srow0) * LDS_STRIDE + scol] = d[it];

    #pragma unroll
    for (int i = 0; i < NCHUNK; ++i) {
        float* cur = (i & 1) ? w1 : w0;
        float* nxt = (i & 1) ? w0 : w1;

        if (i + 1 < NCHUNK) {                      // issue next chunk's loads early
            const int c0 = (i + 1) * CHUNK;
            #pragma unroll
            for (int it = 0; it < 8; ++it)
                d[it] = __builtin_nontemporal_load(
                            (const v4f*)(gbase + (size_t)(it * 2 + srow0) * IN_F + c0));
        }

        // 16 chained WMMAs consume the 64-wide chunk (K += 4 each).
        #pragma unroll
        for (int k = 0; k < CHUNK; k += 4) {
            const v2f av = *(const v2f*)&cur[rowA * LDS_STRIDE + (k + kbase)];
            c = __builtin_amdgcn_wmma_f32_16x16x4_f32(
                    /*neg_a=*/false, av, /*neg_b=*/false, bones,
                    /*c_mod=*/(short)0, c, /*reuse_a=*/false, /*reuse_b=*/false);
        }

        if (i + 1 < NCHUNK) {                      // drain loads into the other buffer
            #pragma unroll
            for (int it = 0; it < 8; ++it)
                *(v4f*)&nxt[(it * 2 + srow0) * LDS_STRIDE + scol] = d[it];
        }
    }

    // C/D layout: VGPR v, lanes 0-15 hold row M=v, lanes 16-31 hold row M=v+8,
    // every N column identical (== row sum). Broadcast-store each row:
    // half-wave writes 256B contiguous per instruction, NT b128 per lane.
    #pragma unroll
    for (int v = 0; v < 8; ++v) {
        const float s  = c[v];
        const int row  = v + ((lane >> 4) << 3);
        v4f val;
        val.x = s; val.y = s; val.z = s; val.w = s;
        float* op = out + (r0 + row) * (size_t)IN_F + ((lane & 15) << 2);
        #pragma unroll
        for (int it = 0; it < 16; ++it)
            __builtin_nontemporal_store(val, (v4f*)(op + it * 64));
    }
}

extern "C" void kernel_launch(void* const* d_in, const int* in_sizes, int n_in,
                              void* d_out, int out_size, void* d_ws, size_t ws_size,
                              hipStream_t stream) {
    const float* x = (const float*)d_in[0];
    float* out = (float*)d_out;

    const int batch = in_sizes[0] / IN_F;                           // 65536
    const int rows_per_block = WAVES_PER_BLOCK * ROWS_PER_WAVE;     // 128
    const int grid = (batch + rows_per_block - 1) / rows_per_block; // 512

    kan_rowsum_wmma<<<grid, 256, 0, stream>>>(x, out);
}